// Router_5171140624829
// MI455X (gfx1250) — compile-verified
//
#include <hip/hip_runtime.h>
#include <math.h>

typedef __attribute__((ext_vector_type(2))) float v2f;
typedef __attribute__((ext_vector_type(8))) float v8f;

#define D_DIM 2048
#define E_DIM 64
#define ROWS  16
#define LDS_STRIDE 68   // pad so per-row reduction lanes hit distinct LDS banks

__global__ __launch_bounds__(128) void router_wmma_kernel(
    const float* __restrict__ x,      // [N, D]
    const float* __restrict__ W,      // [D, E]
    const float* __restrict__ bias_v, // [E]
    float* __restrict__ mask_out,     // [N, E]
    int*   __restrict__ idx_out)      // [N, 2]
{
    __shared__ float s_logits[ROWS * LDS_STRIDE];
    __shared__ float s_val[2][ROWS];
    __shared__ int   s_idx[2][ROWS];

    const int tid  = threadIdx.x;
    const int lane = tid & 31;
    const int wave = tid >> 5;          // 0..3 : which 16-expert column tile
    const int m    = lane & 15;         // A: row-in-tile ; B/C: col-in-tile
    const int half = lane >> 4;         // 0/1 : K phase (A/B), row phase (C)
    const int row_base = blockIdx.x * ROWS;
    const int col = wave * 16 + m;      // expert column 0..63

    // A fragment source: x[row_base + m][k + 2*half + {0,1}]  (aligned float2)
    const float* xrow = x + (size_t)(row_base + m) * D_DIM + 2 * half;
    // B fragment source: W[(k + 2*half + {0,1})][col]
    const float* wcol = W + (size_t)(2 * half) * E_DIM + col;

    v8f acc = {0.f, 0.f, 0.f, 0.f, 0.f, 0.f, 0.f, 0.f};

    for (int k = 0; k < D_DIM; k += 16) {
        // stream-ahead prefetch of the x row (speculative; tail OOB is dropped)
        __builtin_prefetch(xrow + k + 512, 0, 0);
#pragma unroll
        for (int kk = 0; kk < 16; kk += 4) {
            v2f a = *(const v2f*)(xrow + k + kk);
            v2f bf;
            bf.x = wcol[(size_t)(k + kk) * E_DIM];
            bf.y = wcol[(size_t)(k + kk) * E_DIM + E_DIM];
            // D = A(16x4 f32) * B(4x16 f32) + C(16x16 f32)
            acc = __builtin_amdgcn_wmma_f32_16x16x4_f32(
                false, a, false, bf, (short)0, acc, false, false);
        }
    }

    // C/D layout: lane(0..15|16..31) = col, VGPR r = row r | row r+8
    const float bb = bias_v[col];
#pragma unroll
    for (int r = 0; r < 8; ++r)
        s_logits[(r + 8 * half) * LDS_STRIDE + col] = acc[r] + bb;

    __syncthreads();

    // One lane per row: softmax stats + top-2 over 64 experts
    if (tid < ROWS) {
        const float* row = &s_logits[tid * LDS_STRIDE];
        float mx = -INFINITY;
        for (int e = 0; e < E_DIM; ++e) mx = fmaxf(mx, row[e]);
        float s = 0.f;
        for (int e = 0; e < E_DIM; ++e) s += __expf(row[e] - mx);
        float b0 = -INFINITY, b1 = -INFINITY;
        int   i0 = 0, i1 = 0;
        for (int e = 0; e < E_DIM; ++e) {
            float v = row[e];
            if (v > b0)      { b1 = b0; i1 = i0; b0 = v; i0 = e; }
            else if (v > b1) { b1 = v; i1 = e; }
        }
        const float inv = 1.f / s;
        s_val[0][tid] = __expf(b0 - mx) * inv;
        s_val[1][tid] = __expf(b1 - mx) * inv;
        s_idx[0][tid] = i0;
        s_idx[1][tid] = i1;
    }
    __syncthreads();

    // Scatter dispatch mask: 16*64 = 1024 elems, 8 per thread, coalesced
#pragma unroll
    for (int t = 0; t < 8; ++t) {
        const int idx = t * 128 + tid;
        const int r = idx >> 6;
        const int e = idx & 63;
        float v = 0.f;
        if (e == s_idx[0][r])      v = s_val[0][r];
        else if (e == s_idx[1][r]) v = s_val[1][r];
        mask_out[(size_t)(row_base + r) * E_DIM + e] = v;
    }
    if (tid < ROWS) {
        idx_out[(size_t)(row_base + tid) * 2 + 0] = s_idx[0][tid];
        idx_out[(size_t)(row_base + tid) * 2 + 1] = s_idx[1][tid];
    }
}

extern "C" void kernel_launch(void* const* d_in, const int* in_sizes, int n_in,
                              void* d_out, int out_size, void* d_ws, size_t ws_size,
                              hipStream_t stream) {
    const float* x = (const float*)d_in[0];
    const float* W = (const float*)d_in[1];
    const float* b = (const float*)d_in[2];
    const int N = in_sizes[0] / D_DIM;          // 16384

    float* mask_out = (float*)d_out;                       // [N, E] fp32
    int*   idx_out  = (int*)(mask_out + (size_t)N * E_DIM); // [N, 2] int32

    router_wmma_kernel<<<N / ROWS, 128, 0, stream>>>(x, W, b, mask_out, idx_out);
}